// HARCNet_84542136254746
// MI455X (gfx1250) — compile-verified
//
#include <hip/hip_runtime.h>
#include <math.h>

typedef __attribute__((ext_vector_type(2))) float v2f;
typedef __attribute__((ext_vector_type(4))) float v4f;
typedef __attribute__((ext_vector_type(8))) float v8f;

#define WMMA4(a, b, c) \
  __builtin_amdgcn_wmma_f32_16x16x4_f32(false, (a), false, (b), (short)0, (c), false, false)

namespace {
constexpr int D      = 561;   // INPUT_DIM
constexpr int NCLS   = 6;
constexpr int C32    = 32;    // NUM_CAPS * CAP_DIM
constexpr int H      = 64;
constexpr int KCH    = 32;    // K-chunk per double-buffer stage
constexpr int NCHUNK = (D + KCH - 1) / KCH;   // 18
constexpr int S32    = 34;    // stride for 32-wide b64-read tiles (caps/flat)
constexpr int S64    = 68;    // stride for 64-wide b64-read tiles (h1/h2)
constexpr int QS     = 33;    // Q/K/V scratch stride (scalar access only)
constexpr int WAVES  = 4;
constexpr int RPW    = 16;    // rows per wave
constexpr int RPB    = WAVES * RPW;  // 64
constexpr int XFRAG  = 512;   // 16x32 fragment-packed x tile (floats)
constexpr int WFRAG  = 1024;  // 32x32 fragment-packed weight tile (floats)
constexpr int SCR    = 544 + 3 * (16 * QS);        // 2128 floats per-wave scratch
constexpr int LDS_W  = 2 * WAVES * XFRAG;          // 4096
constexpr int LDS_S  = LDS_W + 2 * WFRAG;          // 6144
constexpr int LDS_TOT = LDS_S + WAVES * SCR;       // 14656 floats = 58624 B
}

// Position of element (row r in 0..15, local col c in 0..31) inside a
// fragment-packed 16x32 f32 WMMA operand tile: lane l = r + 16*((c>>1)&1)
// holds its (k,k+1)/(k+2,k+3) pairs for two consecutive k-steps in 16
// contiguous bytes -> one ds_load_b128 feeds two WMMAs.
__device__ __forceinline__ int fragPos(int r, int c) {
  const int s = c >> 2;                     // k-step 0..7
  const int l = r + 16 * ((c >> 1) & 1);    // owning lane
  return ((s >> 1) << 7) + (l << 2) + ((s & 1) << 1) + (c & 1);
}

// Async copy of one 16x32 raw x tile into fragment-packed LDS (16 issues/lane,
// uniform count; out-of-range addresses clamped -> finite data, zeroed weights
// make the products vanish).
__device__ __forceinline__ void fill_x_async(const float* __restrict__ x,
                                             unsigned ldsByteBase, int rowBase,
                                             int kBase, long lastIdx, int lane) {
  const int halfL = lane >> 4;
  const int c0 = lane & 15;
  #pragma unroll
  for (int rp = 0; rp < 8; ++rp) {
    const int rr = rp * 2 + halfL;
    #pragma unroll
    for (int j = 0; j < 2; ++j) {
      const int cl = c0 + 16 * j;                       // local col 0..31
      long idx = (long)(rowBase + rr) * D + kBase + cl; // element index in x
      if (idx > lastIdx) idx = lastIdx;
      const unsigned gOff = (unsigned)idx * 4u;
      const unsigned lOff = ldsByteBase + ((unsigned)fragPos(rr, cl) << 2);
      asm volatile("global_load_async_to_lds_b32 %0, %1, %2"
                   :: "v"(lOff), "v"(gOff), "s"(x) : "memory");
    }
  }
}

// Stage one 32x32 chunk of cap_W (with feature_weights folded in, zero-padded
// past D) into a fragment-packed LDS tile. Branchless.
__device__ __forceinline__ void fill_wc(float* __restrict__ wc,
                                        const float* __restrict__ capW,
                                        const float* __restrict__ fw,
                                        int kBase, int tid) {
  #pragma unroll
  for (int it = 0; it < 8; ++it) {
    const int i = tid + it * 128;        // 0..1023
    const int n = i >> 5;                // output col 0..31
    const int c = i & 31;                // local k
    const int kg = kBase + c;
    const int kgc = (kg < D) ? kg : (D - 1);
    const float msk = (kg < D) ? 1.f : 0.f;
    const float val = capW[n * D + kgc] * fw[kgc] * msk;
    const int tile = n >> 4;
    wc[tile * 512 + fragPos(n & 15, c)] = val;
  }
}

// Branchless B-fragment of the block-diagonal Q/K/V weight:
// Wblk[n][k] = W[n&7][k&7] iff n>>3 == k>>3, else 0.
__device__ __forceinline__ v2f bdiagB(const float* __restrict__ W, int n, int kk) {
  const float w0 = W[(n & 7) * 8 + ( kk      & 7)];
  const float w1 = W[(n & 7) * 8 + ((kk + 1) & 7)];
  v2f b;
  b.x = ((n >> 3) == ( kk      >> 3)) ? w0 : 0.f;
  b.y = ((n >> 3) == ((kk + 1) >> 3)) ? w1 : 0.f;
  return b;
}

__global__ __launch_bounds__(WAVES * 32) void harcnet_fused(
    const float* __restrict__ x,    const float* __restrict__ fw,
    const float* __restrict__ capW, const float* __restrict__ capb,
    const float* __restrict__ qW,   const float* __restrict__ qb,
    const float* __restrict__ kW,   const float* __restrict__ kb,
    const float* __restrict__ vW,   const float* __restrict__ vb,
    const float* __restrict__ f1W,  const float* __restrict__ f1b,
    const float* __restrict__ f2W,  const float* __restrict__ f2b,
    const float* __restrict__ rW,   const float* __restrict__ rb,
    const float* __restrict__ oW,   const float* __restrict__ ob,
    float* __restrict__ out, int nrows)
{
  __shared__ __align__(16) float lds[LDS_TOT];

  const int tid  = threadIdx.x;
  const int wave = tid >> 5;
  const int lane = tid & 31;
  const int half = lane >> 4;
  const int r16  = lane & 15;
  const int ko   = half << 1;
  const int rowBase = blockIdx.x * RPB + wave * RPW;
  const long lastIdx = (long)nrows * D - 1;

  float* scr = lds + LDS_S + wave * SCR;
  const unsigned ldsB = (unsigned)(uintptr_t)(&lds[0]);  // LDS byte offset (low 32 bits)

  const v8f zv = {};

  // ---------------- Phase A: caps = x @ (cap_W*fw)^T, async double-buffered ----------------
  v8f cap0 = zv, cap1 = zv;
  // prologue: stage chunk 0 into buffer 0
  fill_x_async(x, ldsB + (unsigned)(wave * XFRAG) * 4u, rowBase, 0, lastIdx, lane);
  fill_wc(lds + LDS_W, capW, fw, 0, tid);

  for (int ch = 0; ch < NCHUNK; ++ch) {
    const int cur = ch & 1;
    if (ch + 1 < NCHUNK) {
      // stage next chunk into the other buffer while this one computes
      fill_x_async(x, ldsB + (unsigned)((cur ^ 1) * WAVES * XFRAG + wave * XFRAG) * 4u,
                   rowBase, (ch + 1) * KCH, lastIdx, lane);
      fill_wc(lds + LDS_W + (cur ^ 1) * WFRAG, capW, fw, (ch + 1) * KCH, tid);
      asm volatile("s_wait_asynccnt 0x10" ::: "memory");  // only next chunk's 16 in flight
    } else {
      asm volatile("s_wait_asynccnt 0x0" ::: "memory");
    }
    __syncthreads();

    const float* xb = lds + cur * WAVES * XFRAG + wave * XFRAG;
    const float* wc = lds + LDS_W + cur * WFRAG;
    #pragma unroll
    for (int s2 = 0; s2 < 4; ++s2) {
      const v4f av  = *(const v4f*)(xb + s2 * 128 + lane * 4);
      const v4f bv0 = *(const v4f*)(wc + s2 * 128 + lane * 4);
      const v4f bv1 = *(const v4f*)(wc + 512 + s2 * 128 + lane * 4);
      v2f a0;  a0.x  = av[0];  a0.y  = av[1];
      v2f a1;  a1.x  = av[2];  a1.y  = av[3];
      v2f b00; b00.x = bv0[0]; b00.y = bv0[1];
      v2f b01; b01.x = bv0[2]; b01.y = bv0[3];
      v2f b10; b10.x = bv1[0]; b10.y = bv1[1];
      v2f b11; b11.x = bv1[2]; b11.y = bv1[3];
      cap0 = WMMA4(a0, b00, cap0);
      cap1 = WMMA4(a0, b10, cap1);
      cap0 = WMMA4(a1, b01, cap0);
      cap1 = WMMA4(a1, b11, cap1);
    }
    __syncthreads();   // compute done before next iteration refills this buffer
  }

  // caps + bias -> scr[0..544), stride 34
  {
    const float cb0 = capb[r16];
    const float cb1 = capb[r16 + 16];
    #pragma unroll
    for (int v = 0; v < 8; ++v) {
      const int row = v + 8 * half;
      scr[row * S32 + r16]      = cap0[v] + cb0;
      scr[row * S32 + r16 + 16] = cap1[v] + cb1;
    }
  }
  __syncthreads();

  // ---------------- Phase B: Q/K/V via block-diagonal 32x32 GEMMs ----------------
  for (int m = 0; m < 3; ++m) {
    const float* Wm = (m == 0) ? qW : (m == 1) ? kW : vW;
    const float* bm = (m == 0) ? qb : (m == 1) ? kb : vb;
    v8f t0 = zv, t1 = zv;
    #pragma unroll
    for (int s = 0; s < 8; ++s) {
      const int kk = s * 4 + ko;
      const v2f a = *(const v2f*)(scr + r16 * S32 + kk);
      t0 = WMMA4(a, bdiagB(Wm, r16,      kk), t0);
      t1 = WMMA4(a, bdiagB(Wm, r16 + 16, kk), t1);
    }
    float* dst = scr + 544 + m * (16 * QS);
    const float bb = bm[r16 & 7];
    #pragma unroll
    for (int v = 0; v < 8; ++v) {
      const int row = v + 8 * half;
      dst[row * QS + r16]      = t0[v] + bb;
      dst[row * QS + r16 + 16] = t1[v] + bb;
    }
  }
  __syncthreads();

  // ---------------- Phase C: per-row attention (lane -> (row, 2 q-caps)) ----------------
  {
    const float* Qt = scr + 544;
    const float* Kt = Qt + 16 * QS;
    const float* Vt = Kt + 16 * QS;
    const int row = r16;
    float att[2][8];
    #pragma unroll
    for (int qi = 0; qi < 2; ++qi) {
      const int q = half * 2 + qi;
      float s[4];
      #pragma unroll
      for (int kc = 0; kc < 4; ++kc) {
        float d = 0.f;
        #pragma unroll
        for (int e = 0; e < 8; ++e)
          d += Qt[row * QS + q * 8 + e] * Kt[row * QS + kc * 8 + e];
        s[kc] = d * 0.35355339059327373f;   // 1/sqrt(8)
      }
      const float mx = fmaxf(fmaxf(s[0], s[1]), fmaxf(s[2], s[3]));
      float p[4], sum = 0.f;
      #pragma unroll
      for (int kc = 0; kc < 4; ++kc) { p[kc] = __expf(s[kc] - mx); sum += p[kc]; }
      const float inv = 1.f / sum;
      #pragma unroll
      for (int dd = 0; dd < 8; ++dd) {
        float a = 0.f;
        #pragma unroll
        for (int kc = 0; kc < 4; ++kc) a += p[kc] * Vt[row * QS + kc * 8 + dd];
        att[qi][dd] = a * inv;
      }
    }
    #pragma unroll
    for (int qi = 0; qi < 2; ++qi) {
      const int q = half * 2 + qi;
      #pragma unroll
      for (int dd = 0; dd < 8; ++dd)
        scr[row * S32 + q * 8 + dd] = att[qi][dd];   // flat overwrites caps
    }
  }
  __syncthreads();

  // ---------------- Phase D: fc1 + residual GEMMs (shared A-fragments) ----------------
  v8f hA[4], rA[4];
  #pragma unroll
  for (int t = 0; t < 4; ++t) { hA[t] = zv; rA[t] = zv; }
  #pragma unroll
  for (int s = 0; s < 8; ++s) {
    const int kk = s * 4 + ko;
    const v2f a = *(const v2f*)(scr + r16 * S32 + kk);
    #pragma unroll
    for (int t = 0; t < 4; ++t) {
      const int n = t * 16 + r16;
      v2f b1v; b1v.x = f1W[n * C32 + kk]; b1v.y = f1W[n * C32 + kk + 1];
      hA[t] = WMMA4(a, b1v, hA[t]);
      v2f brv; brv.x = rW[n * C32 + kk]; brv.y = rW[n * C32 + kk + 1];
      rA[t] = WMMA4(a, brv, rA[t]);
    }
  }
  // h1 = relu(fc1 + b1) -> scr[544..1632), stride 68 (Q/K/V dead)
  #pragma unroll
  for (int t = 0; t < 4; ++t) {
    const int n = t * 16 + r16;
    const float bb = f1b[n];
    #pragma unroll
    for (int v = 0; v < 8; ++v) {
      const int row = v + 8 * half;
      scr[544 + row * S64 + n] = fmaxf(hA[t][v] + bb, 0.f);
    }
  }
  __syncthreads();

  // ---------------- Phase E: fc2 accumulates onto the residual accumulator ----------------
  #pragma unroll 4
  for (int s = 0; s < 16; ++s) {
    const int kk = s * 4 + ko;
    const v2f a = *(const v2f*)(scr + 544 + r16 * S64 + kk);
    #pragma unroll
    for (int t = 0; t < 4; ++t) {
      const int n = t * 16 + r16;
      v2f b2v; b2v.x = f2W[n * H + kk]; b2v.y = f2W[n * H + kk + 1];
      rA[t] = WMMA4(a, b2v, rA[t]);
    }
  }
  __syncthreads();
  // h2 = relu(fc2 + res + b2 + br) -> scr[544..1632)
  #pragma unroll
  for (int t = 0; t < 4; ++t) {
    const int n = t * 16 + r16;
    const float bb = f2b[n] + rb[n];
    #pragma unroll
    for (int v = 0; v < 8; ++v) {
      const int row = v + 8 * half;
      scr[544 + row * S64 + n] = fmaxf(rA[t][v] + bb, 0.f);
    }
  }
  __syncthreads();

  // ---------------- Phase F: head GEMM [16x64] @ [64x6 padded to 16] ----------------
  const int nO = (r16 < NCLS) ? r16 : 0;       // clamped weight row
  const float om = (r16 < NCLS) ? 1.f : 0.f;   // mask for padded cols
  v8f oA = zv;
  #pragma unroll 4
  for (int s = 0; s < 16; ++s) {
    const int kk = s * 4 + ko;
    const v2f a = *(const v2f*)(scr + 544 + r16 * S64 + kk);
    v2f bo;
    bo.x = oW[nO * H + kk]     * om;
    bo.y = oW[nO * H + kk + 1] * om;
    oA = WMMA4(a, bo, oA);
  }
  if (r16 < NCLS) {
    const float bb = ob[r16];
    #pragma unroll
    for (int v = 0; v < 8; ++v) {
      const int row = rowBase + v + 8 * half;
      if (row < nrows) out[row * NCLS + r16] = oA[v] + bb;
    }
  }
}

extern "C" void kernel_launch(void* const* d_in, const int* in_sizes, int n_in,
                              void* d_out, int out_size, void* d_ws, size_t ws_size,
                              hipStream_t stream) {
  (void)n_in; (void)out_size; (void)d_ws; (void)ws_size;
  const float* x    = (const float*)d_in[0];
  const float* fw   = (const float*)d_in[1];
  const float* capW = (const float*)d_in[2];
  const float* capb = (const float*)d_in[3];
  const float* qW   = (const float*)d_in[4];
  const float* qb   = (const float*)d_in[5];
  const float* kW   = (const float*)d_in[6];
  const float* kb   = (const float*)d_in[7];
  const float* vW   = (const float*)d_in[8];
  const float* vb   = (const float*)d_in[9];
  const float* f1W  = (const float*)d_in[10];
  const float* f1b  = (const float*)d_in[11];
  const float* f2W  = (const float*)d_in[12];
  const float* f2b  = (const float*)d_in[13];
  const float* rW   = (const float*)d_in[14];
  const float* rb   = (const float*)d_in[15];
  const float* oW   = (const float*)d_in[16];
  const float* ob   = (const float*)d_in[17];

  const int nrows = in_sizes[0] / D;
  const int grid  = (nrows + RPB - 1) / RPB;
  harcnet_fused<<<grid, WAVES * 32, 0, stream>>>(
      x, fw, capW, capb, qW, qb, kW, kb, vW, vb,
      f1W, f1b, f2W, f2b, rW, rb, oW, ob,
      (float*)d_out, nrows);
}